// SimpleTTSModel_41077067219191
// MI455X (gfx1250) — compile-verified
//
#include <hip/hip_runtime.h>
#include <math.h>

// SimpleTTSModel: char -> freq embedding, 3307-sample hold, cumsum phase, sin.
// Decomposed: per-char phase base via hierarchical prefix sum (chunk offsets
// computed with V_WMMA_F32_16X16X4_F32 against a strictly-upper triangular
// ones matrix), then a streaming store-bound synthesis kernel using the
// hardware v_sin_f32 (input in revolutions).

#define VOCABX     28
#define CHAR_DUR   3307
#define SRATE      22050.0
#define BATCH      16
#define SEQ        512
#define CHUNK      32
#define NCHUNK     16   // SEQ / CHUNK

typedef float v2f __attribute__((ext_vector_type(2)));
typedef float v8f __attribute__((ext_vector_type(8)));

// ---------------------------------------------------------------------------
// Kernel 1: per-char tables + 32-char chunk sums (Csum is a 16x16 f32 matrix,
// rows = batch row, cols = chunk index).
// ---------------------------------------------------------------------------
__global__ void tts_prep(const int* __restrict__ ids,
                         const float* __restrict__ freq_w,
                         float* __restrict__ segfrac,
                         float* __restrict__ delta,
                         float* __restrict__ amp,
                         float* __restrict__ csum) {
    __shared__ float lds[SEQ];
    const int r = blockIdx.x;        // batch row
    const int s = threadIdx.x;       // char position
    const int id = ids[r * SEQ + s];
    const float f = freq_w[id];      // [vocab,1] -> column 0

    // per-sample phase step, in revolutions
    delta[r * SEQ + s] = (float)((double)f / SRATE);
    // per-char phase advance mod 1 revolution (exact in double)
    double seg = (double)CHAR_DUR * (double)f / SRATE;
    float sf = (float)(seg - floor(seg));
    segfrac[r * SEQ + s] = sf;
    amp[r * SEQ + s] = (id != 0) ? 0.3f : 0.0f;

    lds[s] = sf;
    __syncthreads();
    if (s < NCHUNK) {
        float acc = 0.0f;
        #pragma unroll
        for (int j = 0; j < CHUNK; ++j) acc += lds[s * CHUNK + j];
        csum[r * NCHUNK + s] = acc;   // Csum[row=r][chunk=s]
    }
}

// ---------------------------------------------------------------------------
// Kernel 2: chunk-offset matrix O = Csum x S, S strictly-upper-triangular
// ones => O[r][c] = sum_{c'<c} Csum[r][c']  (exclusive scan per row).
// One wave32; 4 chained V_WMMA_F32_16X16X4_F32 cover K=16.
// ---------------------------------------------------------------------------
__global__ void tts_scan_wmma(const float* __restrict__ csum,
                              float* __restrict__ omat) {
    const int lane = threadIdx.x;          // 0..31, EXEC all ones
    const bool hi  = lane >= 16;
    const int  m   = hi ? lane - 16 : lane; // A-matrix row
    const int  n   = lane & 15;             // B/C column

    v8f c = {};
    #pragma unroll
    for (int step = 0; step < 4; ++step) {
        const int kb = step * 4 + (hi ? 2 : 0);
        v2f a, b;
        // A layout 16x4: lanes0-15 rows M, VGPR0/1 = {K, K+1}; lanes16-31 K+2.
        a.x = csum[m * 16 + kb];
        a.y = csum[m * 16 + kb + 1];
        // B layout 4x16 (mirrors A split): rows of S, S[k][n] = (k < n)
        b.x = (kb     < n) ? 1.0f : 0.0f;
        b.y = (kb + 1 < n) ? 1.0f : 0.0f;
        c = __builtin_amdgcn_wmma_f32_16x16x4_f32(false, a, false, b,
                                                  (short)0, c, false, false);
    }
    // C layout: VGPR v holds M = v (lanes 0-15) / M = v+8 (lanes 16-31)
    const int row0 = hi ? 8 : 0;
    #pragma unroll
    for (int v = 0; v < 8; ++v)
        omat[(row0 + v) * 16 + n] = c[v];
}

// ---------------------------------------------------------------------------
// Kernel 3: per-char phase base (revolutions, fract-reduced):
// base[r][s] = fract( O[r][s/32] + sum_{chunkstart<=j<s} segfrac[r][j] )
// ---------------------------------------------------------------------------
__global__ void tts_base(const float* __restrict__ segfrac,
                         const float* __restrict__ omat,
                         float* __restrict__ base) {
    __shared__ float lds[SEQ];
    const int r = blockIdx.x;
    const int s = threadIdx.x;
    lds[s] = segfrac[r * SEQ + s];
    __syncthreads();
    const int c  = s >> 5;
    const int l0 = c << 5;
    float acc = omat[r * NCHUNK + c];
    for (int j = l0; j < s; ++j) acc += lds[j];
    base[r * SEQ + s] = acc - floorf(acc);
}

// ---------------------------------------------------------------------------
// Kernel 4: streaming synthesis. One block per (row,char) segment of 3307
// samples; coalesced stores; hardware sin (input in revolutions).
// ---------------------------------------------------------------------------
__global__ void tts_synth(const float* __restrict__ base,
                          const float* __restrict__ delta,
                          const float* __restrict__ amp,
                          float* __restrict__ out) {
    const int seg = blockIdx.x;              // 0 .. BATCH*SEQ-1
    const float b = base[seg];
    const float d = delta[seg];
    const float a = amp[seg];
    const unsigned long long o = (unsigned long long)seg * CHAR_DUR;
    for (int k = threadIdx.x; k < CHAR_DUR; k += blockDim.x) {
        float t  = fmaf((float)(k + 1), d, b);   // revolutions, <= ~153
        float fr = t - floorf(t);                // [0,1)
        out[o + k] = a * __builtin_amdgcn_sinf(fr);   // v_sin_f32: sin(2*pi*fr)
    }
}

// ---------------------------------------------------------------------------
// ws layout (floats): segfrac[8192] | delta[8192] | amp[8192] |
//                     csum[256] | omat[256] | base[8192]   (~133 KB)
// ---------------------------------------------------------------------------
extern "C" void kernel_launch(void* const* d_in, const int* in_sizes, int n_in,
                              void* d_out, int out_size, void* d_ws, size_t ws_size,
                              hipStream_t stream) {
    const int*   ids = (const int*)d_in[0];     // [16,512] int32
    const float* fw  = (const float*)d_in[1];   // [28,1]  float32
    float* out = (float*)d_out;                 // [16, 512*3307] float32

    float* ws      = (float*)d_ws;
    float* segfrac = ws;
    float* delta   = ws + BATCH * SEQ;
    float* amp     = ws + 2 * BATCH * SEQ;
    float* csum    = ws + 3 * BATCH * SEQ;
    float* omat    = csum + 256;
    float* basep   = omat + 256;

    tts_prep <<<BATCH, SEQ, 0, stream>>>(ids, fw, segfrac, delta, amp, csum);
    tts_scan_wmma<<<1, 32, 0, stream>>>(csum, omat);
    tts_base <<<BATCH, SEQ, 0, stream>>>(segfrac, omat, basep);
    tts_synth<<<BATCH * SEQ, 256, 0, stream>>>(basep, delta, amp, out);
}